// NADE_66219805769783
// MI455X (gfx1250) — compile-verified
//
#include <hip/hip_runtime.h>
#include <hip/hip_bf16.h>
#include <stdint.h>

typedef __attribute__((ext_vector_type(2))) float        v2f;
typedef __attribute__((ext_vector_type(8))) float        v8f;
typedef __attribute__((ext_vector_type(4))) unsigned int v4u;
typedef __attribute__((ext_vector_type(4))) int          v4i;
typedef __attribute__((ext_vector_type(8))) int          v8i_;

#define NADE_B 64
#define NADE_D 1024
#define NADE_H 1024
#define NWAVES 16                               // 512 threads, wave32
#define TILES_PER_WAVE (NADE_H / 16 / NWAVES)   // 4 h-tiles of 16 per wave
#define NBLK (NADE_D / 16)                      // 64 d-blocks

#if __has_builtin(__builtin_amdgcn_tensor_load_to_lds) && \
    __has_builtin(__builtin_amdgcn_s_wait_tensorcnt)
#define NADE_TDM 1
#else
#define NADE_TDM 0
#endif

// ---------------------------------------------------------------------------
// Tiled 1024x1024 fp32 transpose: dst[c][r] = src[r][c]
// ---------------------------------------------------------------------------
__global__ __launch_bounds__(256) void nade_transpose_1024(
    const float* __restrict__ src, float* __restrict__ dst)
{
    __shared__ float tile[32][33];
    const int lx = threadIdx.x & 31;
    const int ly = threadIdx.x >> 5;
    const int c0 = blockIdx.x * 32;
    const int r0 = blockIdx.y * 32;
#pragma unroll
    for (int i = 0; i < 32; i += 8)
        tile[ly + i][lx] = src[(size_t)(r0 + ly + i) * NADE_D + (c0 + lx)];
    __syncthreads();
#pragma unroll
    for (int i = 0; i < 32; i += 8)
        dst[(size_t)(c0 + ly + i) * NADE_D + (r0 + lx)] = tile[lx][ly + i];
}

#if NADE_TDM
// ---------------------------------------------------------------------------
// Issue one 2D TDM load: global [tile1 rows x tile0 elems, row stride stride0]
// of fp32 into contiguous LDS at lds_addr.  Descriptor per ISA ch.8.
// 6-arg builtin variant: (v4u g0, v8i g1, v4i g2, v4i g3, v8i extra, i32 cpol)
// ---------------------------------------------------------------------------
__device__ __forceinline__ void tdm_load_2d(uint32_t lds_addr, const void* gptr,
                                            uint32_t tile0, uint32_t tile1,
                                            uint32_t stride0)
{
    const uint64_t ga = (uint64_t)(uintptr_t)gptr;
    v4u g0;
    g0.x = 1u;                                          // count=1 (valid), user mode
    g0.y = lds_addr;                                    // LDS byte address
    g0.z = (uint32_t)ga;                                // global_addr[31:0]
    g0.w = (uint32_t)((ga >> 32) & 0x1FFFFFFu)          // global_addr[56:32]
         | (2u << 30);                                  // type = 2 ("image")
    v8i_ g1;
    g1[0] = (int)(2u << 16);                            // data_size = 4 bytes
    g1[1] = (int)((tile0 & 0xFFFFu) << 16);             // tensor_dim0 = tile0
    g1[2] = (int)((tile0 >> 16) | ((tile1 & 0xFFFFu) << 16)); // tensor_dim1 = tile1
    g1[3] = (int)((tile1 >> 16) | (tile0 << 16));       // tile_dim0
    g1[4] = (int)(tile1 & 0xFFFFu);                     // tile_dim1 (tile_dim2 = 0)
    g1[5] = (int)stride0;                               // tensor_dim0_stride lo
    g1[6] = 0;                                          // stride hi / dim1_stride
    g1[7] = 0;
    const v4i  gz4 = {0, 0, 0, 0};                      // groups 2/3: 2D tensor
    const v8i_ gz8 = {0, 0, 0, 0, 0, 0, 0, 0};          // extra descriptor words
    __builtin_amdgcn_tensor_load_to_lds(g0, g1, gz4, gz4, gz8, 0);
}
#endif

// ---------------------------------------------------------------------------
// Main NADE kernel: one workgroup per batch row b.
//   a_base[h] lives in LDS across the whole d loop.
//   d processed in blocks of 16 via fp32 WMMA (16x16x4):
//     step2: a_tile = a_base + Lstrict @ C        (exclusive in-block prefix)
//     step3: diag( relu(a_tile) @ h_W_tile^T )    (accumulated over all h)
//   Weight panels for block db (64KB + 64KB) are DMA'd into double-buffered
//   LDS slabs by the Tensor Data Mover, one block ahead of compute.
// ---------------------------------------------------------------------------
__global__ __launch_bounds__(32 * NWAVES) void nade_main(
    const float* __restrict__ x,      // [B][D]
    const float* __restrict__ in_b,   // [H]
    const float* __restrict__ h_b,    // [D]
    const float* __restrict__ in_Wt,  // [D][H]  (= in_W transposed)
    const float* __restrict__ h_Wt,   // [H][D]  (= h_W transposed)
    float* __restrict__ out)          // [B][D]
{
    extern __shared__ char dynsm[];
    float* slabIn = (float*)dynsm;                            // [2][16][H]
    float* slabHw = (float*)(dynsm + 2u * 16 * NADE_H * 4);   // [2][H][16]

    __shared__ float a_base[NADE_H];               // 4 KB
    __shared__ float xblk[16];
    __shared__ float diagSum[16];
    __shared__ float Atile[NWAVES][16][16];        // 16 KB relu staging

    const int b     = blockIdx.x;
    const int tid   = threadIdx.x;
    const int wave  = tid >> 5;
    const int lane  = tid & 31;
    const int lrow  = lane & 15;
    const int lhalf = lane >> 4;

    for (int h = tid; h < NADE_H; h += blockDim.x) a_base[h] = in_b[h];

#if NADE_TDM
    if (wave == 0) {   // TDM ignores EXEC; one issuing wave per workgroup
        tdm_load_2d((uint32_t)(uintptr_t)slabIn,
                    in_Wt, NADE_H, 16, NADE_H);
        tdm_load_2d((uint32_t)(uintptr_t)slabHw,
                    h_Wt, 16, NADE_H, NADE_D);
        tdm_load_2d((uint32_t)(uintptr_t)(slabIn + 16 * NADE_H),
                    in_Wt + (size_t)16 * NADE_H, NADE_H, 16, NADE_H);
        tdm_load_2d((uint32_t)(uintptr_t)(slabHw + NADE_H * 16),
                    h_Wt + 16, 16, NADE_H, NADE_D);
    }
#endif

    for (int db = 0; db < NBLK; ++db) {
        const int d0  = db * 16;
        const int buf = db & 1;
        float* sIn = slabIn + buf * (16 * NADE_H);   // [j][h]
        float* sHw = slabHw + buf * (NADE_H * 16);   // [h][dd]

#if NADE_TDM
        if (wave == 0) {
            if (db < NBLK - 1) __builtin_amdgcn_s_wait_tensorcnt(2);
            else               __builtin_amdgcn_s_wait_tensorcnt(0);
        }
#else
        {   // fallback: cooperative global->LDS copy of both panels
            const float4* gIn = (const float4*)(in_Wt + (size_t)d0 * NADE_H);
            float4*       lIn = (float4*)sIn;
            for (int i = tid; i < 16 * NADE_H / 4; i += blockDim.x) lIn[i] = gIn[i];
            for (int i = tid; i < NADE_H * 4; i += blockDim.x) {
                const int h = i >> 2, c = (i & 3) * 4;
                *(float4*)(sHw + h * 16 + c) =
                    *(const float4*)(h_Wt + (size_t)h * NADE_D + d0 + c);
            }
        }
#endif
        if (tid < 16) {
            xblk[tid]    = x[(size_t)b * NADE_D + d0 + tid];
            diagSum[tid] = 0.0f;
        }
        __syncthreads();   // panels + xblk visible to all waves

        v8f dacc = {};     // step-3 accumulator (16x16 C/D tile), per wave

        for (int t = 0; t < TILES_PER_WAVE; ++t) {
            const int h0 = wave * (16 * TILES_PER_WAVE) + t * 16;

            // ---- step 2: a_tile = a_base + Lstrict @ C (4x K=4 fp32 WMMA) -
            //      C[j,h] = xblk[j] * in_Wt[d0+j][h], read from TDM slab
            const float ab = a_base[h0 + lrow];
            v8f acc;
#pragma unroll
            for (int r = 0; r < 8; ++r) acc[r] = ab;   // seed accumulator
#pragma unroll
            for (int k4 = 0; k4 < 4; ++k4) {
                v2f Aop, Bop;
#pragma unroll
                for (int v = 0; v < 2; ++v) {
                    const int j = k4 * 4 + v + 2 * lhalf;       // K index
                    Aop[v] = (j < lrow) ? 1.0f : 0.0f;          // Lstrict[M,K]
                    Bop[v] = xblk[j] * sIn[j * NADE_H + h0 + lrow]; // C[K,N]
                }
                acc = __builtin_amdgcn_wmma_f32_16x16x4_f32(
                    false, Aop, false, Bop, (short)0, acc, false, false);
            }

            // pre-relu row 15 (= a_base + sum_{j<15} C[j]) feeds the carry
            const float row15 = acc[7];   // lanes 16..31 hold M=15

            // ---- relu -----------------------------------------------------
#pragma unroll
            for (int r = 0; r < 8; ++r) acc[r] = fmaxf(acc[r], 0.0f);

            // ---- carry a_base across the block: += full column sum --------
            if (lhalf)
                a_base[h0 + lrow] =
                    row15 + xblk[15] * sIn[15 * NADE_H + h0 + lrow];

            // ---- stage relu tile into A-operand layout --------------------
            // C/D layout: vgpr r -> M = r + 8*lhalf, N = lrow
#pragma unroll
            for (int r = 0; r < 8; ++r)
                Atile[wave][r + 8 * lhalf][lrow] = acc[r];
            __builtin_amdgcn_wave_barrier();

            // ---- step 3: dacc += relu(a_tile) @ h_W_tile^T ----------------
#pragma unroll
            for (int k4 = 0; k4 < 4; ++k4) {
                v2f Aop, Bop;
#pragma unroll
                for (int v = 0; v < 2; ++v) {
                    const int k = k4 * 4 + v + 2 * lhalf;       // h offset
                    Aop[v] = Atile[wave][lrow][k];              // A[M=d, K=h]
                    Bop[v] = sHw[(h0 + k) * 16 + lrow];         // B[K, N=d']
                }
                dacc = __builtin_amdgcn_wmma_f32_16x16x4_f32(
                    false, Aop, false, Bop, (short)0, dacc, false, false);
            }
            __builtin_amdgcn_wave_barrier();
        }

        // ---- extract diagonal of dacc and reduce across waves -------------
        // diag(d): d<8  -> vgpr d,   lane d       (0..7)
        //          d>=8 -> vgpr d-8, lane d+16    (24..31)
        {
            const bool owner = (lane < 8) || (lane >= 24);
            const int  d     = (lane < 8) ? lane : (lane - 16);
            const int  r     = (lane < 8) ? lane : (lane - 24);
            float val = 0.0f;
#pragma unroll
            for (int q = 0; q < 8; ++q) val = (r == q) ? dacc[q] : val;
            if (owner) atomicAdd(&diagSum[d], val);   // ds_add_f32
        }
        __syncthreads();

        if (tid < 16) {
            const float z = diagSum[tid] + h_b[d0 + tid];
            out[(size_t)b * NADE_D + d0 + tid] = 1.0f / (1.0f + __expf(-z));
        }
        __syncthreads();   // all waves done with slab[buf] + diagSum

#if NADE_TDM
        if (wave == 0 && db + 2 < NBLK)   // refill freed buffer, 2 blocks ahead
        {
            const int nb = db + 2;
            tdm_load_2d((uint32_t)(uintptr_t)sIn,
                        in_Wt + (size_t)nb * 16 * NADE_H, NADE_H, 16, NADE_H);
            tdm_load_2d((uint32_t)(uintptr_t)sHw,
                        h_Wt + (size_t)nb * 16, 16, NADE_H, NADE_D);
        }
#endif
    }
}

// ---------------------------------------------------------------------------
extern "C" void kernel_launch(void* const* d_in, const int* in_sizes, int n_in,
                              void* d_out, int out_size, void* d_ws, size_t ws_size,
                              hipStream_t stream)
{
    (void)in_sizes; (void)n_in; (void)out_size; (void)ws_size;

    const float* x    = (const float*)d_in[0];   // [B,D]
    const float* in_W = (const float*)d_in[1];   // [H,D]
    const float* in_b = (const float*)d_in[2];   // [H]
    const float* h_W  = (const float*)d_in[3];   // [D,H]
    const float* h_b  = (const float*)d_in[4];   // [D]
    float*       out  = (float*)d_out;           // [B,D]

    float* in_Wt = (float*)d_ws;                          // [D][H], 4 MB
    float* h_Wt  = in_Wt + (size_t)NADE_D * NADE_H;       // [H][D], 4 MB

    dim3 tgrid(32, 32);
    nade_transpose_1024<<<tgrid, 256, 0, stream>>>(in_W, in_Wt);
    nade_transpose_1024<<<tgrid, 256, 0, stream>>>(h_W,  h_Wt);

    // dynamic LDS: double-buffered TDM slabs, 2*(64KB + 64KB) = 256KB
    const size_t dyn_lds = 2u * 16 * NADE_H * 4 + 2u * NADE_H * 16 * 4;
    nade_main<<<dim3(NADE_B), dim3(32 * NWAVES), dyn_lds, stream>>>(
        x, in_b, h_b, in_Wt, h_Wt, out);
}